// KnnModule_17111149707942
// MI455X (gfx1250) — compile-verified
//
#include <hip/hip_runtime.h>
#include <math.h>

// ---------------- problem constants ----------------
#define Q        512          // query rows
#define DDIM     1024         // feature dim
#define NTRAIN   100000       // train rows
#define CHUNK    2048         // train rows per block (chunk)
#define NCHUNK   49           // 49*2048 = 100352 >= 100000
#define MAXK     200
#define CAND     (NCHUNK*MAXK)  // 9800 candidates per row
#define SORT2    16384          // padded size for final sort
#define NCLS     1000
#define TINV     (1.0f/0.07f)

typedef __attribute__((ext_vector_type(2))) float v2f;
typedef __attribute__((ext_vector_type(8))) float v8f;

// =====================================================================
// Kernel 1: fused WMMA GEMM (16 query rows x 2048 train rows) +
//           per-row in-LDS bitonic top-200 selection for this chunk.
// grid = (32 rowTiles, 49 chunks), block = 128 threads (4 waves)
// dynamic LDS: sA 64KB | sS 128KB | sortV 8KB | sortI 8KB  = 208KB
// =====================================================================
extern "C" __global__ void __launch_bounds__(128)
knn_gemm_topk(const float* __restrict__ A, const float* __restrict__ Btr,
              float* __restrict__ candV, int* __restrict__ candI)
{
    extern __shared__ float smem[];
    float* sA   = smem;                    // 16*1024 f32
    float* sS   = sA + 16 * DDIM;          // 16*2048 f32 (sims tile)
    float* sVal = sS + 16 * CHUNK;         // 2048 f32
    int*   sIdx = (int*)(sVal + CHUNK);    // 2048 i32

    const int tid  = threadIdx.x;          // 0..127
    const int lane = tid & 31;
    const int wv   = tid >> 5;             // wave id 0..3
    const int l15  = lane & 15;
    const int half = lane >> 4;            // 0 or 1
    const int rowTile = blockIdx.x;        // 0..31  (x fastest -> L2 reuse of chunk)
    const int chunk   = blockIdx.y;        // 0..48
    const long base   = (long)chunk * CHUNK;

    // ---- stage A tile (16 x 1024) into LDS via float4 ----
    {
        const float4* src = (const float4*)(A + (size_t)rowTile * 16 * DDIM);
        float4* dst = (float4*)sA;
        for (int i = tid; i < 16 * DDIM / 4; i += 128) dst[i] = src[i];
    }
    __syncthreads();

    // ---- WMMA GEMM: each wave owns train-row groups g = wv, wv+4, ... ----
    // A 16x4 fragment: lane(m=l15) supplies K = {2*half, 2*half+1} (v2f)
    // B 4x16 fragment: lane(n=l15) supplies same K pair from train row j
    const float* aRow = sA + (size_t)l15 * DDIM;
    const int ko = 2 * half;
    for (int g = wv; g < CHUNK / 16; g += 4) {
        long j  = base + g * 16 + l15;
        long jc = (j < NTRAIN) ? j : (NTRAIN - 1);   // clamp (masked later)
        const float* bRow = Btr + (size_t)jc * DDIM;

        v8f acc = {0.f, 0.f, 0.f, 0.f, 0.f, 0.f, 0.f, 0.f};
        #pragma unroll 4
        for (int kk = 0; kk < DDIM; kk += 4) {
            v2f af = *(const v2f*)(aRow + kk + ko);
            v2f bf = *(const v2f*)(bRow + kk + ko);
            acc = __builtin_amdgcn_wmma_f32_16x16x4_f32(
                false, af, false, bf, (short)0, acc, false, false);
        }
        // C/D layout: vgpr r, lanes0-15 -> (M=r, N=l15); lanes16-31 -> (M=8+r, N=l15)
        float* sd = sS + g * 16 + l15;
        #pragma unroll
        for (int r = 0; r < 8; ++r)
            sd[(size_t)(half * 8 + r) * CHUNK] = acc[r];
    }
    __syncthreads();

    // ---- per query row: bitonic-sort 2048 (val,idx) desc, emit top-200 ----
    for (int m = 0; m < 16; ++m) {
        for (int i = tid; i < CHUNK; i += 128) {
            long j  = base + i;
            sVal[i] = (j < NTRAIN) ? sS[(size_t)m * CHUNK + i] : -INFINITY;
            sIdx[i] = (int)j;
        }
        __syncthreads();
        for (int ksz = 2; ksz <= CHUNK; ksz <<= 1) {
            for (int jst = ksz >> 1; jst > 0; jst >>= 1) {
                for (int t = tid; t < CHUNK / 2; t += 128) {
                    int i = 2 * t - (t & (jst - 1));
                    int p = i + jst;
                    bool desc = ((i & ksz) == 0);
                    float vi = sVal[i], vp = sVal[p];
                    if (desc ? (vi < vp) : (vi > vp)) {
                        sVal[i] = vp; sVal[p] = vi;
                        int ti = sIdx[i]; sIdx[i] = sIdx[p]; sIdx[p] = ti;
                    }
                }
                __syncthreads();
            }
        }
        const int row = rowTile * 16 + m;
        for (int t = tid; t < MAXK; t += 128) {
            size_t o = ((size_t)row * NCHUNK + chunk) * MAXK + t;
            candV[o] = sVal[t];
            candI[o] = sIdx[t];
        }
        __syncthreads();
    }
}

// =====================================================================
// Kernel 2: per query row — merge 9800 candidates (bitonic sort 16384),
// softmax over top-200, gather labels, cumulative class scatter via
// LDS float atomics; snapshot after k=10,20,100,200.
// grid = 512, block = 256. dynamic LDS: 16384 f32 + 16384 i32 + 1024 f32
// =====================================================================
extern "C" __global__ void __launch_bounds__(256)
knn_finalize(const float* __restrict__ candV, const int* __restrict__ candI,
             const int* __restrict__ labels, float* __restrict__ out)
{
    extern __shared__ float smem[];
    float* sV = smem;                 // 16384
    int*   sI = (int*)(sV + SORT2);   // 16384
    float* sP = (float*)(sI + SORT2); // 1024 (probs[1000] + sum slot at [1000])

    const int row = blockIdx.x;
    const int tid = threadIdx.x;

    for (int i = tid; i < SORT2; i += 256) {
        if (i < CAND) {
            sV[i] = candV[(size_t)row * CAND + i];
            sI[i] = candI[(size_t)row * CAND + i];
        } else {
            sV[i] = -INFINITY;
            sI[i] = 0;
        }
    }
    __syncthreads();

    // bitonic sort descending over 16384
    for (int ksz = 2; ksz <= SORT2; ksz <<= 1) {
        for (int jst = ksz >> 1; jst > 0; jst >>= 1) {
            for (int t = tid; t < SORT2 / 2; t += 256) {
                int i = 2 * t - (t & (jst - 1));
                int p = i + jst;
                bool desc = ((i & ksz) == 0);
                float vi = sV[i], vp = sV[p];
                if (desc ? (vi < vp) : (vi > vp)) {
                    sV[i] = vp; sV[p] = vi;
                    int ti = sI[i]; sI[i] = sI[p]; sI[p] = ti;
                }
            }
            __syncthreads();
        }
    }

    // softmax over the top-200 (temperature 0.07)
    float v0 = sV[0];
    if (tid == 0) sP[NCLS] = 0.f;
    __syncthreads();
    float e = 0.f;
    if (tid < MAXK) {
        e = expf((sV[tid] - v0) * TINV);
        atomicAdd(&sP[NCLS], e);
    }
    __syncthreads();
    float inv = 1.0f / sP[NCLS];

    int lab = 0;
    if (tid < MAXK) lab = labels[sI[tid]];
    __syncthreads();
    if (tid < MAXK) { sV[tid] = e * inv; sI[tid] = lab; }

    for (int c = tid; c < NCLS; c += 256) sP[c] = 0.f;
    __syncthreads();

    // nested-k cumulative accumulation: (10, 20, 100, 200)
    const int bnd[5] = {0, 10, 20, 100, 200};
    #pragma unroll
    for (int s = 0; s < 4; ++s) {
        if (tid >= bnd[s] && tid < bnd[s + 1])
            atomicAdd(&sP[sI[tid]], sV[tid]);      // ds_add_f32
        __syncthreads();
        float* o = out + ((size_t)s * Q + row) * NCLS;
        for (int c = tid; c < NCLS; c += 256) o[c] = sP[c];
        __syncthreads();
    }
}

// =====================================================================
extern "C" void kernel_launch(void* const* d_in, const int* in_sizes, int n_in,
                              void* d_out, int out_size, void* d_ws, size_t ws_size,
                              hipStream_t stream) {
    const float* A      = (const float*)d_in[0];   // 512 x 1024
    const float* Btr    = (const float*)d_in[1];   // 100000 x 1024
    const int*   labels = (const int*)d_in[2];     // 100000
    float* out = (float*)d_out;                    // 4 x 512 x 1000

    // workspace: candV [512 x 9800] f32, then candI [512 x 9800] i32 (~40MB)
    float* candV = (float*)d_ws;
    int*   candI = (int*)((char*)d_ws + (size_t)Q * CAND * sizeof(float));

    const size_t lds1 = (size_t)(16 * DDIM + 16 * CHUNK + CHUNK) * sizeof(float)
                      + (size_t)CHUNK * sizeof(int);                 // 208 KB
    const size_t lds2 = (size_t)SORT2 * sizeof(float)
                      + (size_t)SORT2 * sizeof(int)
                      + 1024 * sizeof(float);                        // 132 KB

    dim3 g1(Q / 16, NCHUNK);   // x = rowTile (fast) so 32 blocks share one chunk in L2
    knn_gemm_topk<<<g1, 128, lds1, stream>>>(A, Btr, candV, candI);
    knn_finalize<<<Q, 256, lds2, stream>>>(candV, candI, labels, out);
}